// DSPEditor_25288767439135
// MI455X (gfx1250) — compile-verified
//
#include <hip/hip_runtime.h>
#include <math.h>

// Problem shape (fixed by the reference)
#define BB 32
#define TT 8192
#define MM 128

#define SCAN_THREADS 1024
#define EPT (TT / SCAN_THREADS)   // 8 elements per thread in the scan
#define FPB 8                     // frames per block in apply (1 wave32 per frame)

// ---------------------------------------------------------------------------
// Kernel 1: per-row segment scan.
// One block per batch row. Stages the label row into LDS via gfx1250 async
// global->LDS DMA, then computes per-frame segment start (cummax) and
// exclusive end (reverse cummin) with a Hillis-Steele scan in LDS.
// Emits meta[b*T+t] = start | (end << 16).
// ---------------------------------------------------------------------------
__global__ __launch_bounds__(SCAN_THREADS)
void dsp_seg_scan_kernel(const int* __restrict__ labels,
                         unsigned int* __restrict__ meta)
{
    // 32 KB of LDS, used twice:
    //  phase A: int labLDS[TT]            (staged label row)
    //  phase B: ushort sarr[TT], earr[TT] (scan arrays)
    __shared__ __align__(16) unsigned short smem[2 * TT];
    unsigned short* sarr = smem;
    unsigned short* earr = smem + TT;
    int* labLDS = (int*)smem;

    const int b   = blockIdx.x;
    const int tid = threadIdx.x;
    const int* row = labels + b * TT;

    // ---- async DMA the whole label row into LDS (b128 = 16B per lane-op) ----
    // TT*4 bytes / 16B = 2048 ops; 1024 threads -> 2 uniform iterations.
    for (int i = tid; i < (TT / 4); i += SCAN_THREADS) {
        unsigned int lds_off = (unsigned int)(unsigned long long)(&labLDS[i * 4]);
        const int* gaddr = row + i * 4;
        asm volatile("global_load_async_to_lds_b128 %0, %1, off"
                     :: "v"(lds_off), "v"(gaddr) : "memory");
    }
    asm volatile("s_wait_asynccnt 0" ::: "memory");
    __syncthreads();

    // ---- boundary flags into registers (LDS about to be overwritten) ----
    int myStart[EPT];
    int myEnd[EPT];
#pragma unroll
    for (int e = 0; e < EPT; ++e) {
        int t  = e * SCAN_THREADS + tid;
        int lt = labLDS[t];
        int lp = (t > 0)      ? labLDS[t - 1] : lt;
        int ln = (t < TT - 1) ? labLDS[t + 1] : lt;
        bool change = (t == 0) || (lt != lp);        // segment start here
        bool last   = (t == TT - 1) || (ln != lt);   // segment ends here
        myStart[e] = change ? t : 0;
        myEnd[e]   = last ? (t + 1) : TT;
    }
    __syncthreads();
#pragma unroll
    for (int e = 0; e < EPT; ++e) {
        int t = e * SCAN_THREADS + tid;
        sarr[t] = (unsigned short)myStart[e];
        earr[t] = (unsigned short)myEnd[e];
    }
    __syncthreads();

    // ---- Hillis-Steele: cummax(start) L->R and cummin(end) R->L, 13 steps ----
    for (int off = 1; off < TT; off <<= 1) {
        unsigned short sv[EPT], ev[EPT];
#pragma unroll
        for (int e = 0; e < EPT; ++e) {
            int t = e * SCAN_THREADS + tid;
            sv[e] = (t >= off)       ? sarr[t - off] : (unsigned short)0;
            ev[e] = (t + off < TT)   ? earr[t + off] : (unsigned short)TT;
        }
        __syncthreads();
#pragma unroll
        for (int e = 0; e < EPT; ++e) {
            int t = e * SCAN_THREADS + tid;
            if (sv[e] > sarr[t]) sarr[t] = sv[e];
            if (ev[e] < earr[t]) earr[t] = ev[e];
        }
        __syncthreads();
    }

#pragma unroll
    for (int e = 0; e < EPT; ++e) {
        int t = e * SCAN_THREADS + tid;
        meta[b * TT + t] = (unsigned int)sarr[t] | ((unsigned int)earr[t] << 16);
    }
}

// ---------------------------------------------------------------------------
// Kernel 2: apply edits. One wave32 per frame; one float4 (4 mel bins) per
// lane -> 512B coalesced row load/store per wave. Lane>=16 <=> bin>=64 for
// the EFFECT low-pass. All label logic is wave-uniform (SALU).
// ---------------------------------------------------------------------------
__global__ __launch_bounds__(FPB * 32)
void dsp_edit_apply_kernel(const float* __restrict__ mel,
                           const int* __restrict__ labels,
                           const unsigned int* __restrict__ meta,
                           float* __restrict__ out)
{
    const int g    = blockIdx.x * FPB + (threadIdx.x >> 5);  // global frame id
    const int lane = threadIdx.x & 31;
    const int b = g / TT;
    const int t = g - b * TT;

    // keep the HBM stream ahead (gfx1250 global_prefetch_b8)
    const int gp = g + 4 * FPB;
    if (gp < BB * TT)
        __builtin_prefetch(mel + (size_t)gp * MM + lane * 4, 0, 1);

    const int label = labels[g];
    const unsigned int m = meta[g];
    const int start   = (int)(m & 0xFFFFu);
    const int end     = (int)(m >> 16);
    const int seg_len = end - start;
    const int pos     = t - start;
    const int dm      = (seg_len - 1) > 1 ? (seg_len - 1) : 1;
    const float frac  = (float)pos / (float)dm;

    size_t srcRow = (size_t)g;
    float gain = 1.0f;
    bool zero = false;
    bool lowp = false;

    if (label == 0 || label == 7) {               // CUT / PAD -> silence
        zero = true;
    } else if (label == 2) {                      // LOOP
        int Lp = seg_len < start ? seg_len : start;
        if (Lp > 0) {
            int s = start - Lp + pos % Lp;
            s = s < 0 ? 0 : (s > TT - 1 ? TT - 1 : s);
            int slab = labels[b * TT + s];
            if (slab == 0 || slab == 7) zero = true;   // source frame was cut/pad
            else srcRow = (size_t)b * TT + (size_t)s;
        }
    } else if (label == 3) {                      // FADE_IN
        gain = frac;
    } else if (label == 4) {                      // FADE_OUT
        gain = 1.0f - frac;
    } else if (label == 6) {                      // TRANSITION
        gain = 1.0f - 0.5f * sinf(3.14159274101257324f * frac);
    } else if (label == 5) {                      // EFFECT (low-pass)
        lowp = true;
    }

    if (lowp && lane >= 16) gain *= 0.3f;         // bins 64..127 * 0.3

    float4 v = make_float4(0.f, 0.f, 0.f, 0.f);
    if (!zero) {
        const float4* src = (const float4*)(mel + srcRow * MM);
        v = src[lane];
        v.x *= gain; v.y *= gain; v.z *= gain; v.w *= gain;
        v.x = fminf(fmaxf(v.x, 0.f), 1.f);
        v.y = fminf(fmaxf(v.y, 0.f), 1.f);
        v.z = fminf(fmaxf(v.z, 0.f), 1.f);
        v.w = fminf(fmaxf(v.w, 0.f), 1.f);
    }
    ((float4*)(out + (size_t)g * MM))[lane] = v;
}

// ---------------------------------------------------------------------------
extern "C" void kernel_launch(void* const* d_in, const int* in_sizes, int n_in,
                              void* d_out, int out_size, void* d_ws, size_t ws_size,
                              hipStream_t stream)
{
    const float* mel    = (const float*)d_in[0];   // (B, T, M) float32
    const int*   labels = (const int*)d_in[1];     // (B, T) int
    float*       out    = (float*)d_out;           // (B, T, M) float32

    unsigned int* meta = (unsigned int*)d_ws;      // B*T*4 = 1 MiB scratch

    dsp_seg_scan_kernel<<<BB, SCAN_THREADS, 0, stream>>>(labels, meta);
    dsp_edit_apply_kernel<<<(BB * TT) / FPB, FPB * 32, 0, stream>>>(mel, labels, meta, out);
}